// SphericalAttentionBlock_9191230014189
// MI455X (gfx1250) — compile-verified
//
#include <hip/hip_runtime.h>
#include <math.h>

#define NLAT 121
#define NLON 240
#define CIN  32
#define HALO 5
#define WIN  11
#define HW   (NLAT * NLON)   // 29040
#define NTILES (HW / 16)     // 1815

typedef __attribute__((ext_vector_type(2))) float v2f;
typedef __attribute__((ext_vector_type(8))) float v8f;

#define PI_D 3.14159265358979323846

__device__ __forceinline__ v8f wmma4(v2f a, v2f b, v8f c) {
    // V_WMMA_F32_16X16X4_F32 : D = A(16x4) * B(4x16) + C(16x16)
    return __builtin_amdgcn_wmma_f32_16x16x4_f32(
        /*neg_a=*/false, a, /*neg_b=*/false, b,
        /*c_mod=*/(short)0, c, /*reuse_a=*/false, /*reuse_b=*/false);
}

// ---------------------------------------------------------------------------
// Kernel 0: device-side geometry (quadrature weights + mask halfwidths),
// computed in double precision to match the numpy reference constants.
// ---------------------------------------------------------------------------
__global__ void sph_geom_kernel(float* __restrict__ quad, int* __restrict__ halfw) {
    const int tid = threadIdx.x;
    if (tid < NLAT) {
        double s = 0.0;
        for (int k = 1; k <= 60; ++k) {
            const double b = (2 * k == 120) ? 1.0 : 2.0;
            s += b * cos(2.0 * PI_D * (double)tid * (double)k / 120.0)
                   / (4.0 * (double)k * (double)k - 1.0);
        }
        double w = (2.0 / 120.0) * (1.0 - s);
        if (tid == 0 || tid == NLAT - 1) w *= 0.5;
        w *= 2.0 * PI_D / (double)NLON;
        quad[tid] = (float)w;
    }
    __syncthreads();
    const double cutoff = 5.0 * PI_D / 120.0 + 1e-12;
    for (int idx = tid; idx < NLAT * WIN; idx += blockDim.x) {
        const int h = idx / WIN, r = idx % WIN;
        const int jr = h + r - HALO;
        int hw = -1;
        if (jr >= 0 && jr < NLAT) {
            const double to = PI_D * (double)h / 120.0;
            const double ti = PI_D * (double)jr / 120.0;
            const double A  = cos(to) * cos(ti);
            const double Bc = sin(to) * sin(ti);
            for (int d = 0; d <= 120; ++d) {          // dist monotone in d
                double cd = A + Bc * cos(2.0 * PI_D * (double)d / (double)NLON);
                cd = cd > 1.0 ? 1.0 : (cd < -1.0 ? -1.0 : cd);
                if (acos(cd) <= cutoff) hw = d; else break;
            }
        }
        halfw[idx] = hw;
    }
}

// ---------------------------------------------------------------------------
// Kernel 1: q/k/v 1x1 projections via WMMA. One wave per 16-point tile.
// x is [C][H*W]; outputs row-major [H*W][C].
// ---------------------------------------------------------------------------
__device__ __forceinline__ void proj16(const v2f a[8], const float* __restrict__ W,
                                       const float* __restrict__ B, float* __restrict__ out,
                                       int p0, int nn, int hh, int koff) {
#pragma unroll
    for (int ct = 0; ct < 2; ++ct) {
        const int co = ct * 16 + nn;
        const float bb = B[co];
        v8f c;
#pragma unroll
        for (int i = 0; i < 8; ++i) c[i] = bb;     // bias in C layout (per-column)
#pragma unroll
        for (int kk = 0; kk < 8; ++kk) {
            v2f b; b.x = W[co * CIN + 4 * kk + koff];
                   b.y = W[co * CIN + 4 * kk + koff + 1];
            c = wmma4(a[kk], b, c);
        }
#pragma unroll
        for (int i = 0; i < 8; ++i)
            out[(p0 + i + hh * 8) * CIN + co] = c[i];
    }
}

__global__ void sph_qkv_kernel(const float* __restrict__ x,
                               const float* __restrict__ qw, const float* __restrict__ qb,
                               const float* __restrict__ kw, const float* __restrict__ kb,
                               const float* __restrict__ vw, const float* __restrict__ vb,
                               float* __restrict__ qo, float* __restrict__ ko,
                               float* __restrict__ vo) {
    const int lane = threadIdx.x;
    const int nn = lane & 15, hh = lane >> 4, koff = hh * 2;
    const int p0 = blockIdx.x * 16;
    const int p  = p0 + nn;
    v2f ax[8];
#pragma unroll
    for (int kk = 0; kk < 8; ++kk) {               // xh[p][c] = x[c*HW + p]
        ax[kk].x = x[(4 * kk + koff) * HW + p];
        ax[kk].y = x[(4 * kk + koff + 1) * HW + p];
    }
    proj16(ax, qw, qb, qo, p0, nn, hh, koff);
    proj16(ax, kw, kb, ko, p0, nn, hh, koff);
    proj16(ax, vw, vb, vo, p0, nn, hh, koff);
}

// ---------------------------------------------------------------------------
// Kernel 2: masked spherical attention, flash-style online softmax.
// One wave per (latitude, 16 output longitudes). s1 = xh + attn(y).
// ---------------------------------------------------------------------------
__global__ void sph_attn_kernel(const float* __restrict__ q,
                                const float* __restrict__ kbuf,
                                const float* __restrict__ vbuf,
                                const float* __restrict__ x,
                                const float* __restrict__ quad,
                                const int* __restrict__ halfw,
                                float* __restrict__ s1) {
    const int tile = blockIdx.x;           // 0..NLAT*15-1
    const int h    = tile / 15;
    const int wo0  = (tile % 15) * 16;
    const int lane = threadIdx.x;
    const int nn = lane & 15, hh = lane >> 4, koff = hh * 2;
    const float scale = 0.17677669529663687f;   // 1/sqrt(32)

    // Q A-fragments (resident all loop long)
    v2f aq[8];
    {
        const float* qr = q + (h * NLON + wo0 + nn) * CIN;
#pragma unroll
        for (int kk = 0; kk < 8; ++kk) {
            aq[kk].x = qr[4 * kk + koff];
            aq[kk].y = qr[4 * kk + koff + 1];
        }
    }

    float Mrun[8], Lrun[8];
    v8f y0 = {}, y1 = {};
#pragma unroll
    for (int i = 0; i < 8; ++i) { Mrun[i] = -1e30f; Lrun[i] = 0.0f; }

    __shared__ float plds[256];            // 16x16 P transpose staging

    for (int r = 0; r < WIN; ++r) {
        const int jr = h + r - HALO;
        const int hw = halfw[h * WIN + r];
        if (hw < 0) continue;              // wave-uniform: EXEC stays all-ones
        const float wt = quad[jr];
        const float* krow = kbuf + jr * NLON * CIN;
        const float* vrow = vbuf + jr * NLON * CIN;
        int nwi = 16 + 2 * hw; if (nwi > NLON) nwi = NLON;
        const int start  = wo0 - hw;
        const int ntile  = (nwi + 15) >> 4;

        for (int t = 0; t < ntile; ++t) {
            const int kidx = t * 16 + nn;          // this lane's key column
            int wik = start + kidx; wik %= NLON; if (wik < 0) wik += NLON;
            const bool inb = (kidx < nwi);

            // K B-fragments
            v2f bk[8];
            const float* kr = krow + wik * CIN;
#pragma unroll
            for (int kk = 0; kk < 8; ++kk) {
                bk[kk].x = kr[4 * kk + koff];
                bk[kk].y = kr[4 * kk + koff + 1];
            }
            v8f s = {};
#pragma unroll
            for (int kk = 0; kk < 8; ++kk) s = wmma4(aq[kk], bk[kk], s);

            // mask + row-max (reduce across the 16 lanes of each half)
            float lg[8]; bool vld[8]; float tmax[8];
#pragma unroll
            for (int i = 0; i < 8; ++i) {
                const int m = i + hh * 8;
                int d = wik - (wo0 + m); d %= NLON; if (d < 0) d += NLON;
                const int md = d < NLON - d ? d : NLON - d;
                vld[i] = inb && (md <= hw);
                lg[i]  = vld[i] ? s[i] * scale : -1e30f;
                float tm = lg[i];
                tm = fmaxf(tm, __shfl_xor(tm, 1));
                tm = fmaxf(tm, __shfl_xor(tm, 2));
                tm = fmaxf(tm, __shfl_xor(tm, 4));
                tm = fmaxf(tm, __shfl_xor(tm, 8));
                tmax[i] = tm;
            }
#pragma unroll
            for (int i = 0; i < 8; ++i) {
                const float newM = fmaxf(Mrun[i], tmax[i]);
                const float corr = expf(Mrun[i] - newM);
                Mrun[i] = newM;
                const float p = vld[i] ? expf(lg[i] - newM) * wt : 0.0f;
                float ps = p;
                ps += __shfl_xor(ps, 1); ps += __shfl_xor(ps, 2);
                ps += __shfl_xor(ps, 4); ps += __shfl_xor(ps, 8);
                Lrun[i] = Lrun[i] * corr + ps;
                y0[i] *= corr; y1[i] *= corr;
                plds[(i + hh * 8) * 16 + nn] = p;   // D-layout -> LDS
            }
            __syncthreads();
            // P A-fragments (transposed read)
            v2f ap[4];
#pragma unroll
            for (int kk = 0; kk < 4; ++kk) {
                ap[kk].x = plds[nn * 16 + 4 * kk + koff];
                ap[kk].y = plds[nn * 16 + 4 * kk + koff + 1];
            }
            // V B-fragments + accumulate (2 col tiles of 16 channels)
#pragma unroll
            for (int kk = 0; kk < 4; ++kk) {
                int k0 = start + t * 16 + 4 * kk + koff;
                int k1 = k0 + 1;
                k0 %= NLON; if (k0 < 0) k0 += NLON;
                k1 %= NLON; if (k1 < 0) k1 += NLON;
                const float* v0 = vrow + k0 * CIN;
                const float* v1 = vrow + k1 * CIN;
                v2f b0; b0.x = v0[nn];      b0.y = v1[nn];
                v2f b1; b1.x = v0[16 + nn]; b1.y = v1[16 + nn];
                y0 = wmma4(ap[kk], b0, y0);
                y1 = wmma4(ap[kk], b1, y1);
            }
            __syncthreads();
        }
    }

    const int p0 = h * NLON + wo0;
#pragma unroll
    for (int i = 0; i < 8; ++i) {
        const int p = p0 + i + hh * 8;
        const float invL = 1.0f / Lrun[i];
        s1[p * CIN + nn]      = y0[i] * invL + x[nn * HW + p];
        s1[p * CIN + 16 + nn] = y1[i] * invL + x[(16 + nn) * HW + p];
    }
}

// ---------------------------------------------------------------------------
// Kernel 3: MLP (fc1 + exact GELU + fc2) with both skips fused; writes [C,H,W].
// ---------------------------------------------------------------------------
__global__ void sph_mlp_kernel(const float* __restrict__ s1,
                               const float* __restrict__ w1, const float* __restrict__ b1,
                               const float* __restrict__ w2, const float* __restrict__ b2,
                               float* __restrict__ out) {
    const int lane = threadIdx.x;
    const int nn = lane & 15, hh = lane >> 4, koff = hh * 2;
    const int p0 = blockIdx.x * 16;
    __shared__ float hl[16 * 64];          // GELU activations, row-major

    v2f as[8];
    const float* sr = s1 + (p0 + nn) * CIN;
#pragma unroll
    for (int kk = 0; kk < 8; ++kk) {
        as[kk].x = sr[4 * kk + koff];
        as[kk].y = sr[4 * kk + koff + 1];
    }
#pragma unroll
    for (int ct = 0; ct < 4; ++ct) {       // 64 hidden channels
        const int co = ct * 16 + nn;
        const float bb = b1[co];
        v8f c;
#pragma unroll
        for (int i = 0; i < 8; ++i) c[i] = bb;
#pragma unroll
        for (int kk = 0; kk < 8; ++kk) {
            v2f b; b.x = w1[co * CIN + 4 * kk + koff];
                   b.y = w1[co * CIN + 4 * kk + koff + 1];
            c = wmma4(as[kk], b, c);
        }
#pragma unroll
        for (int i = 0; i < 8; ++i) {
            const float xv = c[i];
            hl[(i + hh * 8) * 64 + co] = 0.5f * xv * (1.0f + erff(xv * 0.70710678118654752f));
        }
    }
    __syncthreads();
    v2f ah[16];
#pragma unroll
    for (int kk = 0; kk < 16; ++kk) {
        ah[kk].x = hl[nn * 64 + 4 * kk + koff];
        ah[kk].y = hl[nn * 64 + 4 * kk + koff + 1];
    }
#pragma unroll
    for (int ct = 0; ct < 2; ++ct) {
        const int co = ct * 16 + nn;
        const float bb = b2[co];
        v8f c;
#pragma unroll
        for (int i = 0; i < 8; ++i) c[i] = bb;
#pragma unroll
        for (int kk = 0; kk < 16; ++kk) {
            v2f b; b.x = w2[co * 64 + 4 * kk + koff];
                   b.y = w2[co * 64 + 4 * kk + koff + 1];
            c = wmma4(ah[kk], b, c);
        }
#pragma unroll
        for (int i = 0; i < 8; ++i) {
            const int p = p0 + i + hh * 8;
            out[co * HW + p] = c[i] + s1[p * CIN + co];   // skip1 + transpose store
        }
    }
}

// ---------------------------------------------------------------------------
extern "C" void kernel_launch(void* const* d_in, const int* in_sizes, int n_in,
                              void* d_out, int out_size, void* d_ws, size_t ws_size,
                              hipStream_t stream) {
    const float* x   = (const float*)d_in[0];
    const float* qw  = (const float*)d_in[1];
    const float* qb  = (const float*)d_in[2];
    const float* kw  = (const float*)d_in[3];
    const float* kb  = (const float*)d_in[4];
    const float* vw  = (const float*)d_in[5];
    const float* vb  = (const float*)d_in[6];
    const float* w1  = (const float*)d_in[7];
    const float* b1  = (const float*)d_in[8];
    const float* w2  = (const float*)d_in[9];
    const float* b2  = (const float*)d_in[10];
    float* out = (float*)d_out;

    // workspace carve-up
    float* quad  = (float*)d_ws;                           // 121 floats
    int*   halfw = (int*)((char*)d_ws + 1024);             // 1331 ints
    float* qbuf  = (float*)((char*)d_ws + 8192);
    float* kbuf  = qbuf + (size_t)HW * CIN;
    float* vbuf  = kbuf + (size_t)HW * CIN;
    float* s1    = vbuf + (size_t)HW * CIN;

    sph_geom_kernel<<<1, 128, 0, stream>>>(quad, halfw);
    sph_qkv_kernel<<<NTILES, 32, 0, stream>>>(x, qw, qb, kw, kb, vw, vb, qbuf, kbuf, vbuf);
    sph_attn_kernel<<<NLAT * 15, 32, 0, stream>>>(qbuf, kbuf, vbuf, x, quad, halfw, s1);
    sph_mlp_kernel<<<NTILES, 32, 0, stream>>>(s1, w1, b1, w2, b2, out);
}